// IcosphereGradientDirectional_29643864277507
// MI455X (gfx1250) — compile-verified
//
#include <hip/hip_runtime.h>
#include <cstdint>

// Problem constants (LEVEL=10)
#define S_   2048
#define HS_  1024
#define B_   2
#define F_   5

// Tiling: 256 threads/block (8 wave32), tile = 8 rows x 128 cols of output,
// LDS x-tile = 10 rows x 136 cols (1-halo, padded to 16B-aligned column blocks)
#define TI   8
#define TJ   128
#define LROWS (TI + 2)
#define LCOLS (TJ + 8)
#define NUNITS (LROWS * (LCOLS / 4))       // 340 float4 copy units
#define WPL  ((HS_ - 2) * (S_ - 2))        // w_inner plane stride

typedef unsigned int u32x4 __attribute__((ext_vector_type(4)));
typedef int          i32x8 __attribute__((ext_vector_type(8)));
typedef int          i32x4 __attribute__((ext_vector_type(4)));
typedef float        f32x4 __attribute__((ext_vector_type(4)));

// One per-lane async b128 copy unit (clamped halo; used for edge tiles).
__device__ __forceinline__ void stage_unit_async(const float* __restrict__ xbf,
                                                 uint32_t lds_base,
                                                 int i0, int j0, int u) {
  int row = u / (LCOLS / 4);
  int c4  = u % (LCOLS / 4);
  int gi  = i0 - 1 + row;
  gi = gi < 0 ? 0 : (gi > HS_ - 1 ? HS_ - 1 : gi);
  int gj  = j0 - 4 + 4 * c4;
  gj = gj < 0 ? 0 : (gj > S_ - 4 ? S_ - 4 : gj);
  const float* gsrc = xbf + (size_t)gi * S_ + gj;          // 16B aligned
  uint32_t ldst = lds_base + (uint32_t)(row * LCOLS + 4 * c4) * 4u;
  asm volatile("global_load_async_to_lds_b128 %0, %1, off"
               :: "v"(ldst), "v"(gsrc) : "memory");
}

// ---------------------------------------------------------------------------
// Kernel 1: bulk stencil. Interior tiles stage x via one TDM 2D-tile DMA
// (TENSORcnt); edge tiles use per-lane async copies with clamping (ASYNCcnt).
// Every output element is written: stencil inside, zeros elsewhere.
// ---------------------------------------------------------------------------
__global__ __launch_bounds__(256)
void ico_interior_kernel(const float* __restrict__ x,
                         const float* __restrict__ w_inner,
                         float* __restrict__ out) {
  __shared__ float tile[LROWS * LCOLS];

  const int tid = threadIdx.x;
  const int j0  = blockIdx.x * TJ;
  const int i0  = blockIdx.y * TI;
  const int bf  = blockIdx.z;                       // b*F + f
  const float* xbf = x + (size_t)bf * ((size_t)HS_ * S_);

  // Low 32 bits of a flat pointer to __shared__ == wave-relative LDS byte addr.
  const uint32_t lds_base = (uint32_t)(uintptr_t)&tile[0];

  // Interior tiles: the 10x136 halo tile is fully in-bounds -> single TDM op.
  const bool tdm_tile = (blockIdx.y >= 1) && (blockIdx.y <= (HS_ / TI) - 2) &&
                        (blockIdx.x >= 1) && (blockIdx.x <= (S_ / TJ) - 2);

#if __has_builtin(__builtin_amdgcn_tensor_load_to_lds) && \
    __has_builtin(__builtin_amdgcn_s_wait_tensorcnt)
  if (tdm_tile) {
    if (tid < 32) {                                  // one issue per workgroup
      const uint64_t ga =
          (uint64_t)(uintptr_t)(xbf + (size_t)(i0 - 1) * S_ + (j0 - 4));
      u32x4 g0;
      g0.x = 1u;                                     // count=1 (valid D#)
      g0.y = lds_base;                               // lds_addr
      g0.z = (uint32_t)ga;                           // global_addr[31:0]
      g0.w = (uint32_t)((ga >> 32) & 0x01FFFFFFu) | (2u << 30); // addr[56:32], type=2
      i32x8 g1;
      g1[0] = (int)(2u << 16);                       // data_size=4B; no mask/pad/iter
      g1[1] = (int)((uint32_t)(S_ & 0xFFFF) << 16);  // tensor_dim0[15:0]=2048
      g1[2] = (int)((uint32_t)(S_ >> 16) | ((uint32_t)(HS_ & 0xFFFF) << 16)); // dim0 hi, dim1 lo
      g1[3] = (int)((uint32_t)(HS_ >> 16) | ((uint32_t)LCOLS << 16));  // dim1 hi, tile_dim0=136
      g1[4] = (int)LROWS;                            // tile_dim1=10, tile_dim2=0
      g1[5] = (int)S_;                               // tensor_dim0_stride=2048
      g1[6] = 0;
      g1[7] = 0;
      i32x4 z4 = {0, 0, 0, 0};
      i32x8 z8 = {0, 0, 0, 0, 0, 0, 0, 0};
      // amdgpu-toolchain (clang-23) 6-arg form: (g0, g1, g2, g3, extra, cpol)
      __builtin_amdgcn_tensor_load_to_lds(g0, g1, z4, z4, z8, 0);
      __builtin_amdgcn_s_wait_tensorcnt(0);
    }
  } else
#endif
  {
    // Edge tiles (or no-TDM toolchain): per-lane async copies, clamped halo.
    stage_unit_async(xbf, lds_base, i0, j0, tid);
    if (tid < NUNITS - 256) stage_unit_async(xbf, lds_base, i0, j0, tid + 256);
    asm volatile("s_wait_asynccnt 0" ::: "memory");
  }

  // ---- weight loads issued here so their latency hides under the DMA ----
  const int r  = tid >> 5;           // 0..7
  const int c  = tid & 31;           // 0..31
  const int i  = i0 + r;
  const int jb = j0 + 4 * c;
  const bool irow = (i >= 1) && (i <= HS_ - 2);

  float wq[6][4];
  #pragma unroll
  for (int q = 0; q < 4; ++q) {
    const int j = jb + q;
    const bool iv = irow && (j >= 1) && (j <= S_ - 2);
    const int wb = iv ? (i - 1) * (S_ - 2) + (j - 1) : 0;
    #pragma unroll
    for (int k = 0; k < 6; ++k)      // w_inner is single-use: non-temporal
      wq[k][q] = iv ? __builtin_nontemporal_load(&w_inner[k * WPL + wb]) : 0.0f;
  }

  __syncthreads();

  // ---- branchless compute: LDS halo always holds finite x, weight=0 kills
  //      non-interior contributions exactly ----
  const float* T0 = &tile[(r    ) * LCOLS];
  const float* T1 = &tile[(r + 1) * LCOLS];
  const float* T2 = &tile[(r + 2) * LCOLS];

  float o[4][4];
  #pragma unroll
  for (int q = 0; q < 4; ++q) {
    const int ct = 4 * c + 4 + q;              // LDS column of (i, jb+q)
    const float xi = T1[ct];
    o[0][q] = (T0[ct - 1] - xi) * wq[0][q];
    o[1][q] = (T2[ct + 1] - xi) * wq[1][q];
    o[2][q] = (T0[ct    ] - xi) * wq[2][q] + (T1[ct + 1] - xi) * wq[3][q];
    o[3][q] = (T1[ct - 1] - xi) * wq[4][q] + (T2[ct    ] - xi) * wq[5][q];
  }

  const int b = bf / F_, f = bf % F_;
  #pragma unroll
  for (int d = 0; d < 4; ++d) {
    const size_t off = ((((size_t)b * 4 + d) * F_ + f) * HS_ + i) * (size_t)S_ + jb;
    // out is write-once (336 MB): non-temporal keeps x/w resident in L2.
    f32x4 v = {o[d][0], o[d][1], o[d][2], o[d][3]};
    __builtin_nontemporal_store(v, (f32x4*)(out + off));
  }
}

// ---------------------------------------------------------------------------
// Kernel 2: the seven boundary regions (row 0, east column s-1, row hs-1).
// jnp.roll(..., axis=1) rolls the FACE dimension: neighbors from face f∓1 mod 5.
// ---------------------------------------------------------------------------
#define PER_BF ((HS_-1) + 1 + (HS_-2) + 1 + (HS_-2) + 1 + (HS_-1))   // 4093

__global__ __launch_bounds__(256)
void ico_boundary_kernel(const float* __restrict__ x,
                         const float* __restrict__ poles,
                         const float* __restrict__ we,
                         const float* __restrict__ wcn,
                         const float* __restrict__ wne,
                         const float* __restrict__ wnne,
                         const float* __restrict__ wnw,
                         const float* __restrict__ wnww,
                         const float* __restrict__ ww,
                         float* __restrict__ out) {
  const int idx = blockIdx.x * blockDim.x + threadIdx.x;
  if (idx >= B_ * F_ * PER_BF) return;
  const int bf = idx / PER_BF;
  int r = idx % PER_BF;
  const int b = bf / F_, f = bf % F_;
  const int fm = (f + F_ - 1) % F_;
  const int fp = (f + 1) % F_;

  auto X = [&](int ff, int i, int j) -> float {
    return x[(((size_t)b * F_ + ff) * HS_ + i) * (size_t)S_ + j];
  };

  float d0, d1, d2, d3;
  int oi, oj;

  if (r < HS_ - 1) {                                   // ---- east: i=0, j=1..hs-1
    const int t = r;  oi = 0;  oj = 1 + t;
    const float xi = X(f, 0, 1 + t);
    d0 = (X(fm, HS_-1, HS_ + t)     - xi) * we[0*(HS_-1) + t];
    d1 = (X(f, 1, 2 + t)            - xi) * we[1*(HS_-1) + t];
    d2 = (X(fm, HS_-1, HS_ + 1 + t) - xi) * we[2*(HS_-1) + t]
       + (X(f, 0, 2 + t)            - xi) * we[3*(HS_-1) + t];
    d3 = (X(f, 0, t)                - xi) * we[4*(HS_-1) + t]
       + (X(f, 1, 1 + t)            - xi) * we[5*(HS_-1) + t];
  } else if ((r -= HS_ - 1) == 0) {                    // ---- cn: i=0, j=hs
    oi = 0;  oj = HS_;
    const float xi = X(f, 0, HS_);
    d0 = (X(fm, HS_-1, S_-1) - xi) * wcn[0];
    d1 = (X(f, 1, HS_+1)     - xi) * wcn[1];
    d2 = (X(f, 0, HS_+1)     - xi) * wcn[2];
    d3 = (X(f, 0, HS_-1)     - xi) * wcn[3] + (X(f, 1, HS_) - xi) * wcn[4];
  } else if ((r -= 1) < HS_ - 2) {                     // ---- ne: i=0, j=hs+1..s-2
    const int t = r;  oi = 0;  oj = HS_ + 1 + t;
    const float xi = X(f, 0, HS_ + 1 + t);
    d0 = (X(fm, HS_-2 - t, S_-1) - xi) * wne[0*(HS_-2) + t]
       + (X(fm, HS_-1 - t, S_-1) - xi) * wne[1*(HS_-2) + t];
    d1 = (X(f, 1, HS_ + 2 + t)   - xi) * wne[2*(HS_-2) + t]
       + (X(f, 1, HS_ + 1 + t)   - xi) * wne[3*(HS_-2) + t];
    d2 = (X(f, 0, HS_ + 2 + t)   - xi) * wne[4*(HS_-2) + t];
    d3 = (X(f, 0, HS_ + t)       - xi) * wne[5*(HS_-2) + t];
  } else if ((r -= HS_ - 2) == 0) {                    // ---- nne: i=0, j=s-1
    oi = 0;  oj = S_ - 1;
    const float xi = X(f, 0, S_-1);
    d0 = (X(fm, 0, S_-1) - xi) * wnne[0] + (X(fm, 1, S_-1) - xi) * wnne[1];
    d1 = (X(fp, 0, S_-1) - xi) * wnne[2] + (X(f, 1, S_-1)  - xi) * wnne[3];
    d2 = (poles[b*2 + 1] - xi) * wnne[4];
    d3 = (X(f, 0, S_-2)  - xi) * wnne[5];
  } else if ((r -= 1) < HS_ - 2) {                     // ---- nw: i=1..hs-2, j=s-1
    const int t = r;  oi = 1 + t;  oj = S_ - 1;
    const float xi = X(f, 1 + t, S_-1);
    d0 = (X(f, t, S_-2)           - xi) * wnw[0*(HS_-2) + t];
    d1 = (X(fp, 0, S_-2 - t)      - xi) * wnw[1*(HS_-2) + t];
    d2 = (X(f, t, S_-1)           - xi) * wnw[2*(HS_-2) + t]
       + (X(fp, 0, S_-1 - t)      - xi) * wnw[3*(HS_-2) + t];
    d3 = (X(f, 1 + t, S_-2)       - xi) * wnw[4*(HS_-2) + t]
       + (X(f, 2 + t, S_-1)       - xi) * wnw[5*(HS_-2) + t];
  } else if ((r -= HS_ - 2) == 0) {                    // ---- nww: i=hs-1, j=s-1
    oi = HS_ - 1;  oj = S_ - 1;
    const float xi = X(f, HS_-1, S_-1);
    d0 = (X(f, HS_-2, S_-2) - xi) * wnww[0];
    d1 = (X(fp, 0, HS_)     - xi) * wnww[1];
    d2 = (X(f, HS_-2, S_-1) - xi) * wnww[2] + (X(fp, 0, HS_+1) - xi) * wnww[3];
    d3 = (X(f, HS_-1, S_-2) - xi) * wnww[4] + (X(fp, 0, HS_-1) - xi) * wnww[5];
  } else {                                             // ---- west: i=hs-1, j=hs..s-2
    const int t = r - 1;  oi = HS_ - 1;  oj = HS_ + t;
    const float xi = X(f, HS_-1, HS_ + t);
    d0 = (X(f, HS_-2, HS_-1 + t) - xi) * ww[0*(HS_-1) + t];
    d1 = (X(fp, 0, 1 + t)        - xi) * ww[1*(HS_-1) + t];
    d2 = (X(f, HS_-2, HS_ + t)   - xi) * ww[2*(HS_-1) + t]
       + (X(f, HS_-1, HS_+1 + t) - xi) * ww[3*(HS_-1) + t];
    d3 = (X(f, HS_-1, HS_-1 + t) - xi) * ww[4*(HS_-1) + t]
       + (X(fp, 0, t)            - xi) * ww[5*(HS_-1) + t];
  }

  const float dv[4] = {d0, d1, d2, d3};
  #pragma unroll
  for (int d = 0; d < 4; ++d) {
    out[((((size_t)b * 4 + d) * F_ + f) * HS_ + oi) * (size_t)S_ + oj] = dv[d];
  }
}

// ---------------------------------------------------------------------------
extern "C" void kernel_launch(void* const* d_in, const int* in_sizes, int n_in,
                              void* d_out, int out_size, void* d_ws, size_t ws_size,
                              hipStream_t stream) {
  const float* x       = (const float*)d_in[0];
  const float* poles   = (const float*)d_in[1];
  const float* w_inner = (const float*)d_in[2];
  const float* w_east  = (const float*)d_in[3];
  const float* w_cn    = (const float*)d_in[4];
  const float* w_ne    = (const float*)d_in[5];
  const float* w_nne   = (const float*)d_in[6];
  const float* w_nw    = (const float*)d_in[7];
  const float* w_nww   = (const float*)d_in[8];
  const float* w_west  = (const float*)d_in[9];
  float* out = (float*)d_out;

  // Bulk: writes every output element (interior stencil, zeros elsewhere).
  dim3 grid(S_ / TJ, HS_ / TI, B_ * F_);           // 16 x 128 x 10 blocks
  ico_interior_kernel<<<grid, 256, 0, stream>>>(x, w_inner, out);

  // Boundaries: overwrite the seven special regions (ordered on same stream).
  const int total = B_ * F_ * PER_BF;
  ico_boundary_kernel<<<(total + 255) / 256, 256, 0, stream>>>(
      x, poles, w_east, w_cn, w_ne, w_nne, w_nw, w_nww, w_west, out);
}